// MACE_16561393893485
// MI455X (gfx1250) — compile-verified
//
#include <hip/hip_runtime.h>
#include <math.h>

#define N_NODESC 8192
#define N_EDGESC 131072
#define CCH 64
#define LCOMP 16
#define NBASIS 8
#define OUTD 128
#define RMAXF 5.0f
#define INV_AVG 0.0625f
#define NROWS (N_NODESC * LCOMP)          // 131072 rows of the node feature matrix
#define NH ((size_t)NROWS * CCH)          // floats per h buffer

typedef __attribute__((ext_vector_type(2))) float v2f;
typedef __attribute__((ext_vector_type(8))) float v8f;

// ---------------------------------------------------------------- utilities
__global__ void zero_f32(float* __restrict__ p, size_t n) {
  size_t i = (size_t)blockIdx.x * blockDim.x + threadIdx.x;
  if (i < n) p[i] = 0.0f;
}

__global__ void init_h(const float* __restrict__ emb, const int* __restrict__ species,
                       float* __restrict__ h) {
  size_t i = (size_t)blockIdx.x * blockDim.x + threadIdx.x;
  if (i >= NH) return;
  int c = (int)(i & 63);
  int l = (int)((i >> 6) & 15);
  int n = (int)(i >> 10);
  h[i] = (l == 0) ? emb[species[n] * CCH + c] : 0.0f;
}

// ------------------------------------------------- per-edge geometry (once)
__global__ void geom(const float* __restrict__ pos, const int* __restrict__ snd,
                     const int* __restrict__ rcv, float* __restrict__ Y,
                     float* __restrict__ RB) {
  int e = blockIdx.x * blockDim.x + threadIdx.x;
  if (e >= N_EDGESC) return;
  int s = snd[e], r = rcv[e];
  float vx = pos[3 * r + 0] - pos[3 * s + 0];
  float vy = pos[3 * r + 1] - pos[3 * s + 1];
  float vz = pos[3 * r + 2] - pos[3 * s + 2];
  float rr = sqrtf(vx * vx + vy * vy + vz * vz);
  float rinv = 1.0f / fmaxf(rr, 1e-6f);
  float x = vx * rinv, y = vy * rinv, z = vz * rinv;
  const float s3 = 1.7320508075688772f;   // sqrt(3)
  const float c1 = 0.7905694150420949f;   // 0.5*sqrt(2.5)
  const float c2 = 3.8729833462074170f;   // sqrt(15)
  const float c3 = 0.6123724356957945f;   // 0.5*sqrt(1.5)
  float x2 = x * x, y2 = y * y, z2 = z * z;
  float* Ye = Y + (size_t)e * 16;
  Ye[0]  = 1.0f;
  Ye[1]  = x;  Ye[2] = y;  Ye[3] = z;
  Ye[4]  = s3 * x * y;
  Ye[5]  = s3 * y * z;
  Ye[6]  = 0.5f * (3.0f * z2 - 1.0f);
  Ye[7]  = s3 * x * z;
  Ye[8]  = 0.5f * s3 * (x2 - y2);
  Ye[9]  = c1 * y * (3.0f * x2 - y2);
  Ye[10] = c2 * x * y * z;
  Ye[11] = c3 * y * (5.0f * z2 - 1.0f);
  Ye[12] = 0.5f * z * (5.0f * z2 - 3.0f);
  Ye[13] = c3 * x * (5.0f * z2 - 1.0f);
  Ye[14] = 0.5f * c2 * z * (x2 - y2);
  Ye[15] = c1 * x * (x2 - 3.0f * y2);
  // Bessel radial basis * soft envelope (envelope folded in)
  float rs = fmaxf(rr, 1e-6f);
  float u  = fminf(rr / RMAXF, 1.0f - 1e-6f);
  float u2 = u * u;
  float env = (rr < RMAXF) ? expf(2.0f * u2 / (u2 - 1.0f)) : 0.0f;
  float pref = sqrtf(2.0f / RMAXF) / rs * env;
  const float PI = 3.14159265358979323846f;
  float* rbe = RB + (size_t)e * NBASIS;
#pragma unroll
  for (int n = 1; n <= NBASIS; n++)
    rbe[n - 1] = pref * sinf((float)n * PI * rs / RMAXF);
}

// ------------------------------------- message + scatter (fp32 L2 atomics)
// 64 threads per edge, one channel per thread.
__global__ void edge_msg(const float* __restrict__ h, const float* __restrict__ Y,
                         const float* __restrict__ RB, const float* __restrict__ wrad,
                         const int* __restrict__ snd, const int* __restrict__ rcv,
                         float* __restrict__ agg) {
  size_t gid = (size_t)blockIdx.x * blockDim.x + threadIdx.x;
  int e = (int)(gid >> 6);
  int c = (int)(gid & 63);
  if (e >= N_EDGESC) return;
  int s = snd[e], r = rcv[e];
  const float* rbe = RB + (size_t)e * NBASIS;
  float w = 0.0f;
#pragma unroll
  for (int b = 0; b < NBASIS; b++) w = fmaf(rbe[b], wrad[b * CCH + c], w);
  const float* hs = h + (size_t)s * (LCOMP * CCH);
  const float* Ye = Y + (size_t)e * 16;
  float h0w = hs[c] * w;
  float* ar = agg + (size_t)r * (LCOMP * CCH) + c;
#pragma unroll
  for (int l = 0; l < LCOMP; l++) {
    float m = fmaf(Ye[l], h0w, hs[l * CCH + c] * w);
    atomicAdd(ar + l * CCH, m);   // global_atomic_add_f32, no return
  }
}

// ------------------- node update: h = (agg/16)@w_mix + h@w_skip  (fp32 WMMA)
// 4 waves per block, wave w owns 16-column tile w*16; B frags live in regs.
__global__ void __launch_bounds__(128) node_update(
    const float* __restrict__ agg, const float* __restrict__ h,
    const float* __restrict__ wmix, const float* __restrict__ wskip,
    float* __restrict__ hout) {
  int lane = threadIdx.x & 31;
  int wave = threadIdx.x >> 5;
  int nb   = wave * 16;
  int rowInTile = lane & 15;
  int koff = (lane >> 4) << 1;       // lanes 0-15 -> K 0/1, lanes 16-31 -> K 2/3
  int col  = lane & 15;

  v2f bMix[16], bSkip[16];
#pragma unroll
  for (int kk = 0; kk < 16; kk++) {
    int k0 = kk * 4 + koff;
    bMix[kk].x  = wmix[(k0    ) * CCH + nb + col];
    bMix[kk].y  = wmix[(k0 + 1) * CCH + nb + col];
    bSkip[kk].x = wskip[(k0    ) * CCH + nb + col];
    bSkip[kk].y = wskip[(k0 + 1) * CCH + nb + col];
  }

  const int nRowTiles = NROWS / 16;  // 8192
  for (int rt = blockIdx.x; rt < nRowTiles; rt += gridDim.x) {
    size_t rowBase = (size_t)rt * 16 + rowInTile;
    const float* arow = agg + rowBase * CCH;
    const float* hrow = h   + rowBase * CCH;
    v8f acc = {};
#pragma unroll
    for (int kk = 0; kk < 16; kk++) {
      int k0 = kk * 4 + koff;
      v2f aA = *(const v2f*)(arow + k0);
      aA.x *= INV_AVG; aA.y *= INV_AVG;      // fold 1/AVG_NEIGH into A frag
      acc = __builtin_amdgcn_wmma_f32_16x16x4_f32(false, aA, false, bMix[kk],
                                                  (short)0, acc, false, false);
      v2f aH = *(const v2f*)(hrow + k0);
      acc = __builtin_amdgcn_wmma_f32_16x16x4_f32(false, aH, false, bSkip[kk],
                                                  (short)0, acc, false, false);
    }
    int m0 = (lane < 16) ? 0 : 8;
    float* orow = hout + ((size_t)rt * 16) * CCH + nb + col;
#pragma unroll
    for (int i = 0; i < 8; i++) orow[(m0 + i) * CCH] = acc[i];
  }
}

// -------------------------- readout: out[:, t*128:] = h[:,0,:] @ w_read[t]
// 8 waves per block, wave w owns 16-column tile of the 128 output columns.
__global__ void __launch_bounds__(256) readout(
    const float* __restrict__ h, const float* __restrict__ wread,
    float* __restrict__ out, int colOff) {
  int lane = threadIdx.x & 31;
  int wave = threadIdx.x >> 5;
  int nb   = wave * 16;
  int rowInTile = lane & 15;
  int koff = (lane >> 4) << 1;
  int col  = lane & 15;

  v2f bR[16];
#pragma unroll
  for (int kk = 0; kk < 16; kk++) {
    int k0 = kk * 4 + koff;
    bR[kk].x = wread[(k0    ) * OUTD + nb + col];
    bR[kk].y = wread[(k0 + 1) * OUTD + nb + col];
  }

  const int nRowTiles = N_NODESC / 16;  // 512
  for (int rt = blockIdx.x; rt < nRowTiles; rt += gridDim.x) {
    // l=0 slice: row stride through h is LCOMP*CCH floats
    const float* arow = h + (size_t)(rt * 16 + rowInTile) * (LCOMP * CCH);
    v8f acc = {};
#pragma unroll
    for (int kk = 0; kk < 16; kk++) {
      int k0 = kk * 4 + koff;
      v2f a = *(const v2f*)(arow + k0);
      acc = __builtin_amdgcn_wmma_f32_16x16x4_f32(false, a, false, bR[kk],
                                                  (short)0, acc, false, false);
    }
    int m0 = (lane < 16) ? 0 : 8;
    float* orow = out + ((size_t)rt * 16) * (2 * OUTD) + colOff + nb + col;
#pragma unroll
    for (int i = 0; i < 8; i++) orow[(m0 + i) * (2 * OUTD)] = acc[i];
  }
}

// ---------------------------------------------------------------- launcher
extern "C" void kernel_launch(void* const* d_in, const int* in_sizes, int n_in,
                              void* d_out, int out_size, void* d_ws, size_t ws_size,
                              hipStream_t stream) {
  const float* positions     = (const float*)d_in[0];
  const float* species_embed = (const float*)d_in[1];
  const float* w_rad         = (const float*)d_in[2];  // [2,8,64]
  const float* w_mix         = (const float*)d_in[3];  // [2,64,64]
  const float* w_skip        = (const float*)d_in[4];  // [2,64,64]
  const float* w_read        = (const float*)d_in[5];  // [2,64,128]
  const int*   species       = (const int*)d_in[6];
  const int*   senders       = (const int*)d_in[7];
  const int*   receivers     = (const int*)d_in[8];
  float* out = (float*)d_out;

  float* hA  = (float*)d_ws;
  float* hB  = hA + NH;
  float* agg = hB + NH;
  float* Yb  = agg + NH;
  float* RBb = Yb + (size_t)N_EDGESC * 16;

  geom<<<(N_EDGESC + 255) / 256, 256, 0, stream>>>(positions, senders, receivers, Yb, RBb);
  init_h<<<(int)(NH / 256), 256, 0, stream>>>(species_embed, species, hA);

  const float* hcur = hA;
  float* hnext = hB;
  for (int t = 0; t < 2; t++) {
    zero_f32<<<(int)(NH / 256), 256, 0, stream>>>(agg, NH);
    edge_msg<<<(N_EDGESC * 64) / 256, 256, 0, stream>>>(
        hcur, Yb, RBb, w_rad + (size_t)t * NBASIS * CCH, senders, receivers, agg);
    node_update<<<1024, 128, 0, stream>>>(
        agg, hcur, w_mix + (size_t)t * CCH * CCH, w_skip + (size_t)t * CCH * CCH, hnext);
    readout<<<128, 256, 0, stream>>>(
        hnext, w_read + (size_t)t * CCH * OUTD, out, t * OUTD);
    const float* tmp = hnext; hnext = (float*)hcur; hcur = tmp;
  }
}